// MixtureCrossattention_4045859193702
// MI455X (gfx1250) — compile-verified
//
#include <hip/hip_runtime.h>
#include <hip/hip_bf16.h>
#include <math.h>

// MixtureCrossattention on gfx1250:
//  - per-batch fused attention, all intermediates in LDS (no 680MB A matrix in HBM)
//  - fp32 WMMA (v_wmma_f32_16x16x4_f32) for both GEMMs, softmax in fp32
typedef __attribute__((ext_vector_type(2))) float v2f;
typedef __attribute__((ext_vector_type(8))) float v8f;

#define ES 292   // e-matrix LDS row stride (288 + pad), floats
#define PS 612   // P scratch row stride (576 + pad), floats; 2448B, 16B aligned
#define NW 4     // waves per workgroup

__global__ __launch_bounds__(128)
void mca_attn_kernel(const float* __restrict__ x1,
                     const float* __restrict__ x2,
                     float* __restrict__ out, int B)
{
    extern __shared__ float smem[];
    float* e1s = smem;                 // [16][ES]
    float* e2s = e1s + 16 * ES;        // [16][ES]
    float* ps  = e2s + 16 * ES;        // [NW][16][PS]  per-wave P^T scratch

    const int b    = blockIdx.x;
    const int tid  = threadIdx.x;
    const int wv   = tid >> 5;
    const int lane = tid & 31;
    const int c    = lane & 15;        // column index (query/key/d depending on role)
    const int hi   = lane >> 4;        // K-half selector for WMMA frags

    // ---------------- patch-embed gather: x[32,12,12] -> e[16][288] ----------------
    {
        const float* srcs[2] = { x1 + (size_t)b * 4608, x2 + (size_t)b * 4608 };
        float*       dsts[2] = { e1s, e2s };
        for (int m = 0; m < 2; ++m) {
            const float* src = srcs[m];
            float*       dst = dsts[m];
            for (int idx = tid; idx < 4608; idx += 128) {
                int ch  = idx / 144;
                int rem = idx - ch * 144;
                int row = rem / 12;
                int col = rem - row * 12;
                int h = row / 3, p1 = row - h * 3;
                int w = col / 3, p2 = col - w * 3;
                int r = h * 4 + w;                    // token index 0..15
                int f = (p1 * 3 + p2) * 32 + ch;      // feature index 0..287
                dst[r * ES + f] = src[idx];
            }
        }
    }
    __syncthreads();

    float* myrow = ps + (size_t)(wv * 16 + c) * PS;   // this lane's P^T row (query col c)

    // 36 units = 2 passes x 18 query tiles; 9 per wave (uniform trip count)
    for (int u = wv; u < 36; u += NW) {
        const int p  = u / 18;                 // 0 -> x3 (Q=e2), 1 -> x4 (Q=e1)
        const int qt = u - p * 18;
        const float* eQ   = p ? e1s : e2s;
        const float* eKlo = p ? e2s : e1s;     // keys i <  288 (V scale 0.7)
        const float* eKhi = p ? e1s : e2s;     // keys i >= 288 (V scale 0.3)
        const int j = qt * 16 + c;             // global query feature (0..287)

        // ---- Q fragments (B operand), 4 k-steps over the 16 token dims ----
        v2f qb[4];
        #pragma unroll
        for (int kk = 0; kk < 4; ++kk) {
            int d0 = 4 * kk + 2 * hi;
            v2f q; q[0] = eQ[d0 * ES + j]; q[1] = eQ[(d0 + 1) * ES + j];
            qb[kk] = q;
        }

        // ---- scores S[key, query] = K^T Q : 36 key tiles, track column max ----
        float mx = -3.0e38f;
        for (int t = 0; t < 36; ++t) {
            const float* eK = (t < 18) ? eKlo : eKhi;
            const int ic = ((t < 18) ? t * 16 : (t - 18) * 16) + c;   // key col in eK
            v8f acc = {};
            #pragma unroll
            for (int kk = 0; kk < 4; ++kk) {
                int d0 = 4 * kk + 2 * hi;
                v2f a; a[0] = eK[d0 * ES + ic]; a[1] = eK[(d0 + 1) * ES + ic];
                acc = __builtin_amdgcn_wmma_f32_16x16x4_f32(
                          false, a, false, qb[kk], (short)0, acc, false, false);
            }
            // D layout: acc[v] = S[16t + v + 8*hi][c]  -> store as P^T row c
            float* dst = myrow + t * 16 + 8 * hi;
            *(float4*)(dst)     = make_float4(acc[0], acc[1], acc[2], acc[3]);
            *(float4*)(dst + 4) = make_float4(acc[4], acc[5], acc[6], acc[7]);
            mx = fmaxf(mx, fmaxf(fmaxf(fmaxf(acc[0], acc[1]), fmaxf(acc[2], acc[3])),
                                 fmaxf(fmaxf(acc[4], acc[5]), fmaxf(acc[6], acc[7]))));
        }
        mx = fmaxf(mx, __shfl_xor(mx, 16, 32));   // lane pair shares query column

        // ---- softmax exp pass: fold V scale (0.7 / 0.3) into P, sum denominator ----
        asm volatile("s_wait_dscnt 0" ::: "memory");
        float sum = 0.0f;
        for (int t = 0; t < 36; ++t) {
            const float vs = (t < 18) ? 0.7f : 0.3f;
            float* ptr = myrow + t * 16 + 8 * hi;
            float4 a4 = *(float4*)(ptr);
            float4 b4 = *(float4*)(ptr + 4);
            float q0 = __expf(a4.x - mx), q1 = __expf(a4.y - mx);
            float q2 = __expf(a4.z - mx), q3 = __expf(a4.w - mx);
            float q4 = __expf(b4.x - mx), q5 = __expf(b4.y - mx);
            float q6 = __expf(b4.z - mx), q7 = __expf(b4.w - mx);
            sum += ((q0 + q1) + (q2 + q3)) + ((q4 + q5) + (q6 + q7));
            *(float4*)(ptr)     = make_float4(q0 * vs, q1 * vs, q2 * vs, q3 * vs);
            *(float4*)(ptr + 4) = make_float4(q4 * vs, q5 * vs, q6 * vs, q7 * vs);
        }
        sum += __shfl_xor(sum, 16, 32);
        const float inv = 1.0f / sum;

        // ---- O[d, query] = V @ P : K = 576, 144 chained WMMA k-steps ----
        asm volatile("s_wait_dscnt 0" ::: "memory");
        v8f o = {};
        #pragma unroll 4
        for (int ks = 0; ks < 144; ++ks) {
            int i0 = 4 * ks + 2 * hi;                       // key index pair i0, i0+1
            const float* eV = (ks < 72) ? eKlo : eKhi;      // uniform across wave
            int ii = i0 - ((ks < 72) ? 0 : 288);
            v2f a = *(const v2f*)(eV + c * ES + ii);        // V[d=c][ii..ii+1] (scale in P)
            v2f bb = *(const v2f*)(myrow + i0);             // P[i0..i0+1][c]
            o = __builtin_amdgcn_wmma_f32_16x16x4_f32(
                    false, a, false, bb, (short)0, o, false, false);
        }

        // ---- un-patch scatter: O[d][j] -> out[pass][b][ch][h*3+p1][w*3+p2] ----
        const int ch  = j & 31;
        const int p12 = j >> 5;
        const int p1i = p12 / 3, p2i = p12 - p1i * 3;
        float* ob = out + (p ? (size_t)B * 4608 : (size_t)0)
                        + (size_t)b * 4608 + ch * 144 + p1i * 12 + p2i;
        #pragma unroll
        for (int v = 0; v < 8; ++v) {
            int d = v + 8 * hi;                // token index
            int h = d >> 2, w = d & 3;
            ob[h * 36 + w * 3] = o[v] * inv;   // deferred softmax normalization
        }
    }
}

extern "C" void kernel_launch(void* const* d_in, const int* in_sizes, int n_in,
                              void* d_out, int out_size, void* d_ws, size_t ws_size,
                              hipStream_t stream) {
    const float* x1 = (const float*)d_in[0];
    const float* x2 = (const float*)d_in[1];
    float* out = (float*)d_out;
    const int B = in_sizes[0] / 4608;                     // 512
    const size_t shmem = (size_t)(2 * 16 * ES + NW * 16 * PS) * sizeof(float); // ~190 KB
    mca_attn_kernel<<<B, 128, shmem, stream>>>(x1, x2, out, B);
}